// GCN_17660905521700
// MI455X (gfx1250) — compile-verified
//
#include <hip/hip_runtime.h>

typedef __attribute__((ext_vector_type(2))) float v2f;
typedef __attribute__((ext_vector_type(8))) float v8f;

// ---------------------------------------------------------------------------
// Utility kernels
// ---------------------------------------------------------------------------
__global__ void zero_f(float* __restrict__ p, long long n) {
    long long i = (long long)blockIdx.x * blockDim.x + threadIdx.x;
    if (i < n) p[i] = 0.0f;
}

__global__ void degree_accum(const int* __restrict__ src, const int* __restrict__ dst,
                             float* __restrict__ dsrc, float* __restrict__ ddst, int E) {
    int e = blockIdx.x * blockDim.x + threadIdx.x;
    if (e < E) {
        atomicAdd(dsrc + src[e], 1.0f);
        atomicAdd(ddst + dst[e], 1.0f);
    }
}

// in-place: x = rsqrt(max(x,1))   (applied to both norm arrays, contiguous 2N)
__global__ void deg_to_norm(float* __restrict__ p, int n) {
    int i = blockIdx.x * blockDim.x + threadIdx.x;
    if (i < n) {
        float d = p[i];
        d = d < 1.0f ? 1.0f : d;
        p[i] = rsqrtf(d);
    }
}

// ---------------------------------------------------------------------------
// Dual GEMM with WMMA f32 16x16x4:
//   P[m, n] = sum_k (X[m,k] * ns[m]) * Wm[k, n]
//   D[m, n] = sum_k  X[m,k]          * Lm[k, n]
// Block = 256 threads = 8 waves. Block handles one 16-row strip; wave w handles
// column tile w. All reads of X complete before the barrier, so D may alias X
// (in-place skip-connection output).
//
// K / LDX / LDW are template constants so every load address in the fully
// unrolled K-loop folds into the instruction's 24-bit immediate offset
// (no per-iteration 64-bit address arithmetic).
//
// Out-of-range columns: WMMA output column n depends only on B column n, and
// columns >= Nout are never stored, so their B values are don't-care. We clamp
// the load column (avoids OOB reads) instead of branching -> EXEC stays
// all-ones for WMMA with no per-iteration mask manipulation.
// ---------------------------------------------------------------------------
template <int K, int LDX, int LDW>
__global__ __launch_bounds__(256) void gemm_dual(
    const float* __restrict__ X, const float* __restrict__ Wm,
    const float* __restrict__ Lm, const float* __restrict__ ns,
    float* __restrict__ P, float* __restrict__ D,
    int M, int Nout, int ldp, int ldd)
{
    const int lane = threadIdx.x & 31;
    const int wave = threadIdx.x >> 5;
    const int ln   = lane & 15;          // A row sel / B,C col sel
    const int hh   = lane >> 4;          // K-half / C row-half
    const int rowBase = blockIdx.x * 16;
    const int n0      = wave * 16;
    const int tiles16 = (Nout + 15) >> 4;
    const bool active = wave < tiles16;

    v8f accW = {0,0,0,0,0,0,0,0};
    v8f accL = {0,0,0,0,0,0,0,0};

    if (active) {
        int arow = rowBase + ln;
        if (arow > M - 1) arow = M - 1;          // clamp keeps EXEC all-ones
        int ncol = n0 + ln;
        if (ncol > Nout - 1) ncol = Nout - 1;    // clamp: don't-care columns
        const float nsv  = ns[arow];
        const float* xrow = X + (size_t)arow * LDX + 2 * hh;
        const float* wcol = Wm + (size_t)(2 * hh) * LDW + ncol;
        const float* lcol = Lm + (size_t)(2 * hh) * LDW + ncol;

        #pragma unroll
        for (int k0 = 0; k0 < K; k0 += 4) {
            v2f a  = *(const v2f*)(xrow + k0);           // b64 load, K-pair
            v2f as; as.x = a.x * nsv; as.y = a.y * nsv;  // fold norm_src into A
            v2f bw; bw.x = wcol[k0 * LDW]; bw.y = wcol[(k0 + 1) * LDW];
            v2f bl; bl.x = lcol[k0 * LDW]; bl.y = lcol[(k0 + 1) * LDW];
            accW = __builtin_amdgcn_wmma_f32_16x16x4_f32(
                false, as, false, bw, (short)0, accW, false, false);
            accL = __builtin_amdgcn_wmma_f32_16x16x4_f32(
                false, a,  false, bl, (short)0, accL, false, false);
        }
    }

    __syncthreads();   // all X reads done before D (possibly aliasing X) is written

    if (active) {
        const int ncol = n0 + ln;
        if (ncol < Nout) {
            #pragma unroll
            for (int r = 0; r < 8; ++r) {
                int row = rowBase + r + 8 * hh;
                if (row < M) {
                    P[(size_t)row * ldp + ncol] = accW[r];
                    D[(size_t)row * ldd + ncol] = accL[r];
                }
            }
        }
    }
}

// ---------------------------------------------------------------------------
// SpMM aggregation: Mb[dst[e], :] += P[src[e], :]   (one wave per edge)
// P and Mb are packed with row stride F. L2-resident, atomic f32 adds.
// ---------------------------------------------------------------------------
__global__ __launch_bounds__(256) void scatter_add(
    const float* __restrict__ P, const int* __restrict__ src,
    const int* __restrict__ dst, float* __restrict__ Mb, int E, int F)
{
    int e = blockIdx.x * (blockDim.x >> 5) + (threadIdx.x >> 5);
    if (e >= E) return;
    int lane = threadIdx.x & 31;
    int s = src[e], d = dst[e];
    const float* p = P + (size_t)s * F;
    float*       m = Mb + (size_t)d * F;
    if (F == 128) {
        float4 v = *(const float4*)(p + lane * 4);   // b128 gather (L2 hit)
        float* mp = m + lane * 4;
        atomicAdd(mp + 0, v.x);
        atomicAdd(mp + 1, v.y);
        atomicAdd(mp + 2, v.z);
        atomicAdd(mp + 3, v.w);
    } else {
        for (int f = lane; f < F; f += 32) atomicAdd(m + f, p[f]);
    }
}

// ---------------------------------------------------------------------------
// y[i, f] = Mb[i, f] * nd[i] + D[i, f] (+ bias[f])   -- grid = N rows
// ---------------------------------------------------------------------------
__global__ void combine(const float* __restrict__ Mb, const float* __restrict__ nd,
                        const float* __restrict__ D, const float* __restrict__ bias,
                        float* __restrict__ Y, int F, int ldd)
{
    int i = blockIdx.x;
    int f = threadIdx.x;
    if (f < F) {
        float v = Mb[(size_t)i * F + f] * nd[i] + D[(size_t)i * ldd + f];
        if (bias) v += bias[f];
        Y[(size_t)i * F + f] = v;
    }
}

// ---------------------------------------------------------------------------
// Launch
// ---------------------------------------------------------------------------
extern "C" void kernel_launch(void* const* d_in, const int* in_sizes, int n_in,
                              void* d_out, int out_size, void* d_ws, size_t ws_size,
                              hipStream_t stream)
{
    const float* feat = (const float*)d_in[0];
    const int*   src  = (const int*)d_in[1];
    const int*   dst  = (const int*)d_in[2];
    const float* W0   = (const float*)d_in[3];
    const float* W1   = (const float*)d_in[4];
    const float* W2   = (const float*)d_in[5];
    const float* b2   = (const float*)d_in[6];
    const float* L0   = (const float*)d_in[7];
    const float* L1   = (const float*)d_in[8];
    const float* L2   = (const float*)d_in[9];
    float* out = (float*)d_out;

    const int FIN = 128, H = 128;
    const int N = in_sizes[0] / FIN;
    const int E = in_sizes[1];
    const int C = in_sizes[6];           // 40

    // workspace layout (floats)
    float* nsrc = (float*)d_ws;          // N
    float* ndst = nsrc + N;              // N   (contiguous with nsrc)
    float* bufP = ndst + N;              // N*H
    float* bufH = bufP + (size_t)N * H;  // N*H : D target / layer activations
    float* bufM = bufH + (size_t)N * H;  // N*H : aggregation target

    const int mt = (N + 15) / 16;        // GEMM row tiles
    const int ebw = (E + 7) / 8;         // scatter blocks (8 waves/block)

    // degrees -> norms
    zero_f<<<(2LL * N + 255) / 256, 256, 0, stream>>>(nsrc, 2LL * N);
    degree_accum<<<(E + 255) / 256, 256, 0, stream>>>(src, dst, nsrc, ndst, E);
    deg_to_norm<<<(2 * N + 255) / 256, 256, 0, stream>>>(nsrc, 2 * N);

    // ---- Layer 0: h = agg((feat*ns)@W0)*ndst + feat@L0
    gemm_dual<128, 128, 128><<<mt, 256, 0, stream>>>(feat, W0, L0, nsrc, bufP, bufH,
                                                     N, H, H, H);
    zero_f<<<((long long)N * H + 255) / 256, 256, 0, stream>>>(bufM, (long long)N * H);
    scatter_add<<<ebw, 256, 0, stream>>>(bufP, src, dst, bufM, E, H);
    combine<<<N, 128, 0, stream>>>(bufM, ndst, bufH, nullptr, bufH, H, H);

    // ---- Layer 1 (D written in place over bufH; barrier in gemm_dual makes it safe)
    gemm_dual<128, 128, 128><<<mt, 256, 0, stream>>>(bufH, W1, L1, nsrc, bufP, bufH,
                                                     N, H, H, H);
    zero_f<<<((long long)N * H + 255) / 256, 256, 0, stream>>>(bufM, (long long)N * H);
    scatter_add<<<ebw, 256, 0, stream>>>(bufP, src, dst, bufM, E, H);
    combine<<<N, 128, 0, stream>>>(bufM, ndst, bufH, nullptr, bufH, H, H);

    // ---- Layer 2: C=40 outputs (P/M packed stride C; D in place stride H)
    gemm_dual<128, 128, 40><<<mt, 256, 0, stream>>>(bufH, W2, L2, nsrc, bufP, bufH,
                                                    N, C, C, H);
    zero_f<<<((long long)N * C + 255) / 256, 256, 0, stream>>>(bufM, (long long)N * C);
    scatter_add<<<ebw, 256, 0, stream>>>(bufP, src, dst, bufM, E, C);
    combine<<<N, 64, 0, stream>>>(bufM, ndst, bufH, b2, out, C, H);
}